// SMap3x3_29927332118729
// MI455X (gfx1250) — compile-verified
//
#include <hip/hip_runtime.h>
#include <stdint.h>

#define TW 32
#define TH 8
#define HWD 34   // halo tile width  (TW+2)
#define HHT 10   // halo tile height (TH+2)
#define LST 36   // LDS row stride (in pixels)
#define IN_H 318
#define IN_W 318
#define PH 320
#define PW 320
#define NIMG 16  // B*C = 2*8

// Phase-1: argmin over the 9 projected-coordinate costs, first-occurrence ties.
__device__ __forceinline__ int smap_ind(int gi, int gj, float z, float kx, float ky,
                                        float c00, float c01, float c02,
                                        float c10, float c11, float c12,
                                        float off_h, float off_w, float sy, float sx) {
  float xs[3], ys[3];
  bool okx[3], oky[3];
#pragma unroll
  for (int t = 0; t < 3; ++t) {
    const int jj = gj + t - 1;
    okx[t] = ((unsigned)jj < (unsigned)PW);
    xs[t] = ((float)jj - off_w + 0.5f) * sx;
  }
#pragma unroll
  for (int s = 0; s < 3; ++s) {
    const int ii = gi + s - 1;
    oky[s] = ((unsigned)ii < (unsigned)PH);
    ys[s] = ((float)ii - off_h + 0.5f) * sy;
  }
  float best = 3.4e38f;
  int ind = 0, k = 0;
#pragma unroll
  for (int s = 0; s < 3; ++s) {
#pragma unroll
    for (int t = 0; t < 3; ++t) {
      float qx = 0.0f, qy = 0.0f;
      if (oky[s] && okx[t]) {   // unfold zero-pads the coordinate image
        const float zx = z * xs[t];
        const float zy = z * ys[s];
        qx = c00 * zx + c01 * zy + c02 * z;
        qy = c10 * zx + c11 * zy + c12 * z;
      }
      const float cost = fabsf(kx - qx) + fabsf(ky - qy);
      if (cost < best) { best = cost; ind = k; }  // strict <: first-occurrence min
      ++k;
    }
  }
  return ind;
}

__global__ __launch_bounds__(256, 1) void smap3x3_kernel(
    const float* __restrict__ x,
    const float* __restrict__ cam,
    const int* __restrict__ p_oh,
    const int* __restrict__ p_ow,
    float* __restrict__ out)
{
  // AoS pixel tile: (kx, ky, z, r) per halo pixel, plus the phase-1 index plane.
  __shared__ float4 sPix[HHT * LST];
  __shared__ int    sId [HHT * LST];

  const int bx = blockIdx.x, by = blockIdx.y, bc = blockIdx.z;
  const int tx = threadIdx.x, ty = threadIdx.y;
  const int tid = ty * TW + tx;

  // 3x3 camera-matrix inverse, f32 adjugate (exact for this matrix class; the
  // downstream use is a cost comparison with ~3e-3 slot separation, so ulp-level
  // differences vs jnp.linalg.inv cannot flip the argmin).
  const float a00 = cam[0], a01 = cam[1], a02 = cam[2];
  const float a10 = cam[3], a11 = cam[4], a12 = cam[5];
  const float a20 = cam[6], a21 = cam[7], a22 = cam[8];
  const float m00 = a11 * a22 - a12 * a21;
  const float m01 = a10 * a22 - a12 * a20;
  const float m02 = a10 * a21 - a11 * a20;
  const float det = a00 * m00 - a01 * m01 + a02 * m02;
  const float id = 1.0f / det;
  const float c00 = m00 * id;
  const float c01 = (a02 * a21 - a01 * a22) * id;
  const float c02 = (a01 * a12 - a02 * a11) * id;
  const float c10 = (a12 * a20 - a10 * a22) * id;
  const float c11 = (a00 * a22 - a02 * a20) * id;
  const float c12 = (a02 * a10 - a00 * a12) * id;

  const int oh = p_oh[0], ow = p_ow[0];
  const float off_h = (float)((PH - oh) / 2);
  const float off_w = (float)((PW - ow) / 2);
  const float sy = 318.0f / (float)oh;   // IMG_SHAPE[0] / orig_h
  const float sx = 318.0f / (float)ow;   // IMG_SHAPE[1] / orig_w

  const float* img = x + (size_t)bc * 4u * IN_H * IN_W;
  const int gi0 = by * TH - 1;   // padded-grid row of halo li=0
  const int gj0 = bx * TW - 1;

  // Block whose whole halo maps inside the raw 318x318 input.
  const bool interior = (bx >= 1) && (bx <= 8) && (by >= 1) && (by <= 38);

  if (interior) {
    // gfx1250 async data mover: stage the 4 channel planes straight into the
    // AoS LDS tile (per-lane dword writes at slot*16 + ch*4).
#pragma unroll
    for (int ch = 0; ch < 4; ++ch) {
      const float* src = img + (size_t)ch * IN_H * IN_W;
      for (int p = tid; p < HHT * HWD; p += 256) {
        const int li = p / HWD;
        const int lj = p - li * HWD;
        const float* ga = src + (size_t)(gi0 + li - 1) * IN_W + (gj0 + lj - 1);
        const unsigned ldsa =
            (unsigned)(uintptr_t)((float*)(sPix + (li * LST + lj)) + ch);
        const unsigned long long g = (unsigned long long)(uintptr_t)ga;
        asm volatile("global_load_async_to_lds_b32 %0, %1, off"
                     :: "v"(ldsa), "v"(g)
                     : "memory");
      }
    }
    asm volatile("s_wait_asynccnt 0" ::: "memory");
    // Phase 1 for the pixels this thread staged (same p mapping -> no barrier yet).
    for (int p = tid; p < HHT * HWD; p += 256) {
      const int li = p / HWD;
      const int lj = p - li * HWD;
      const int o = li * LST + lj;
      const float4 v = sPix[o];
      sId[o] = smap_ind(gi0 + li, gj0 + lj, v.z, v.x, v.y,
                        c00, c01, c02, c10, c11, c12, off_h, off_w, sy, sx);
    }
  } else {
    // Border tiles: guarded scalar loads, zero-fill outside (exactly equals the
    // reference's pad/shift-fct semantics since r=0 pixels contribute nothing).
    for (int p = tid; p < HHT * HWD; p += 256) {
      const int li = p / HWD;
      const int lj = p - li * HWD;
      const int yi = gi0 + li - 1;
      const int xj = gj0 + lj - 1;
      float4 v = make_float4(0.0f, 0.0f, 0.0f, 0.0f);
      if (((unsigned)yi < (unsigned)IN_H) && ((unsigned)xj < (unsigned)IN_W)) {
        const size_t off = (size_t)yi * IN_W + xj;
        v.x = img[off];
        v.y = img[off + (size_t)IN_H * IN_W];
        v.z = img[off + 2u * (size_t)IN_H * IN_W];
        v.w = img[off + 3u * (size_t)IN_H * IN_W];
      }
      const int o = li * LST + lj;
      sPix[o] = v;
      sId[o] = smap_ind(gi0 + li, gj0 + lj, v.z, v.x, v.y,
                        c00, c01, c02, c10, c11, c12, off_h, off_w, sy, sx);
    }
  }
  __syncthreads();

  // Phase 2: gather the 9 incoming slots, argmin by sender z (first occurrence),
  // accumulate nrm, override to center slot when msum <= 0.5.
  const int ly = ty + 1, lx = tx + 1;
  const int cbase = ly * LST + lx;
  float msum = 0.0f;
  float bestz = 1.0e7f;
  int bestk = 0;
#pragma unroll
  for (int k = 0; k < 9; ++k) {
    const int dy = 1 - k / 3;
    const int dx = 1 - (k % 3);
    const int o = cbase + dy * LST + dx;
    const float4 v = sPix[o];
    const int idn = sId[o];
    const bool hit = (v.w > 0.5f) && (idn == k);
    const float zi = (hit && v.z > 0.0f) ? v.z : 1.0e7f;
    if (zi < bestz) { bestz = zi; bestk = k; }
    float nr = 0.0f;
    if (hit) nr = v.w;
    else if (k == 4 && !(v.w > 0.5f)) nr = v.w;  // center-slot override of new_r_mask
    msum += nr;
  }
  const int idx = (msum > 0.5f) ? bestk : 4;
  const int dy = 1 - idx / 3;
  const int dx = 1 - (idx % 3);
  const int o = cbase + dy * LST + dx;
  const float4 v = sPix[o];
  const int idn = sId[o];
  const bool on = (v.w > 0.5f) && (idn == idx);   // weights_b one-hot
  const float wx = on ? v.x : 0.0f;
  const float wy = on ? v.y : 0.0f;
  const float wz = on ? v.z : 0.0f;
  float wr = 0.0f;
  if (on) wr = v.w;
  else if (idx == 4 && !(v.w > 0.5f)) wr = v.w;

  const int i = by * TH + ty;
  const int j = bx * TW + tx;
  const size_t plane = (size_t)PH * PW;
  const size_t ob = (size_t)bc * 4u * plane + (size_t)i * PW + j;
  out[ob]             = wx;
  out[ob + plane]     = wy;
  out[ob + 2 * plane] = wz;
  out[ob + 3 * plane] = wr;
}

extern "C" void kernel_launch(void* const* d_in, const int* in_sizes, int n_in,
                              void* d_out, int out_size, void* d_ws, size_t ws_size,
                              hipStream_t stream) {
  (void)in_sizes; (void)n_in; (void)d_ws; (void)ws_size; (void)out_size;
  const float* x  = (const float*)d_in[0];
  const float* cm = (const float*)d_in[1];
  const int* oh   = (const int*)d_in[2];
  const int* ow   = (const int*)d_in[3];
  dim3 grid(PW / TW, PH / TH, NIMG);   // 10 x 40 x 16
  dim3 block(TW, TH, 1);               // 256 threads = 8 wave32
  smap3x3_kernel<<<grid, block, 0, stream>>>(x, cm, oh, ow, (float*)d_out);
}